// Transformer_F_52510270161093
// MI455X (gfx1250) — compile-verified
//
#include <hip/hip_runtime.h>

// Transformer_F: only column 64 of Z evolves; columns 0..63 are invariant.
// Per batch b:  K = Zh[:512]^T Zh[:512]  (WMMA f32 16x16x4),
//               u = Zh[:512]^T v0, then 4-step 64-dim recursion,
//               v_final = v0 + Zh @ S, write Z with col 64 replaced.

typedef __attribute__((ext_vector_type(2))) float v2f;
typedef __attribute__((ext_vector_type(8))) float v8f;

#define B_ROWS 513   // N+1
#define B_COLS 65    // D+1
#define DH     64    // D
#define NSEQ   512   // n
#define NLAYER 4

__global__ __launch_bounds__(256)
void tf_col64_kernel(const float* __restrict__ Z,
                     const float* __restrict__ allparam,
                     const float* __restrict__ gamma,
                     const float* __restrict__ alpha,
                     float* __restrict__ out)
{
    __shared__ float v0[B_ROWS];      // Z[b,:,64]
    __shared__ float Kmat[DH * DH];   // Gram matrix
    __shared__ float uvec[DH];
    __shared__ float tvec[DH];
    __shared__ float Svec[DH];
    __shared__ float ktv[DH];
    __shared__ float cacc[NLAYER];    // per-layer diag means

    const int b    = blockIdx.x;
    const int tid  = threadIdx.x;
    const int lane = tid & 31;
    const int wave = tid >> 5;        // 8 waves of 32 (wave32)
    const int half = lane >> 4;       // 0: lanes 0-15, 1: lanes 16-31
    const int l16  = lane & 15;

    const float* __restrict__ Zb = Z   + (size_t)b * B_ROWS * B_COLS;
    float*       __restrict__ Ob = out + (size_t)b * B_ROWS * B_COLS;

    // ---- stage 0: zero LDS accumulators ------------------------------------
    if (tid < NLAYER) cacc[tid] = 0.f;
    if (tid < DH) { uvec[tid] = 0.f; tvec[tid] = 0.f; Svec[tid] = 0.f; }
    __syncthreads();

    // ---- stage 1: load v0 (column 64) and c_i = mean(diag(Q_i)) ------------
    for (int n = tid; n < B_ROWS; n += 256) v0[n] = Zb[(size_t)n * B_COLS + DH];
    {   // 256 threads cover all 4 layers x 64 diag elements
        const int li = tid >> 6, k = tid & 63;
        float dv = allparam[(size_t)(2 * li + 1) * DH * DH + (size_t)k * (DH + 1)];
        atomicAdd(&cacc[li], dv * (1.0f / DH));
    }
    __syncthreads();

    // ---- stage 2: K = Zh[:512]^T @ Zh[:512] via v_wmma_f32_16x16x4_f32 -----
    // 16 output tiles (4x4 grid of 16x16); 2 tiles per wave.
    for (int t = wave; t < 16; t += 8) {
        const int ti = t >> 2, tj = t & 3;
        // A[i,kk] = Zh[m+kk, 16*ti+i]; ISA layout: lanes0-15 hold K=0/1
        // in VGPR0/1, lanes16-31 hold K=2/3.
        const float* pA = Zb + (size_t)(2 * half) * B_COLS + ti * 16 + l16;
        const float* pB = Zb + (size_t)(2 * half) * B_COLS + tj * 16 + l16;
        v8f acc0 = {};
        v8f acc1 = {};
        for (int m = 0; m < NSEQ; m += 8) {
            v2f a0, b0, a1, b1;
            a0.x = pA[(size_t)m       * B_COLS]; a0.y = pA[(size_t)(m + 1) * B_COLS];
            b0.x = pB[(size_t)m       * B_COLS]; b0.y = pB[(size_t)(m + 1) * B_COLS];
            a1.x = pA[(size_t)(m + 4) * B_COLS]; a1.y = pA[(size_t)(m + 5) * B_COLS];
            b1.x = pB[(size_t)(m + 4) * B_COLS]; b1.y = pB[(size_t)(m + 5) * B_COLS];
            acc0 = __builtin_amdgcn_wmma_f32_16x16x4_f32(
                       false, a0, false, b0, (short)0, acc0, false, false);
            acc1 = __builtin_amdgcn_wmma_f32_16x16x4_f32(
                       false, a1, false, b1, (short)0, acc1, false, false);
        }
        // D layout: VGPR g -> row g (lanes 0-15) / row g+8 (lanes 16-31), col = lane%16
        #pragma unroll
        for (int g = 0; g < 8; ++g)
            Kmat[(ti * 16 + g + 8 * half) * DH + tj * 16 + l16] = acc0[g] + acc1[g];
    }

    // ---- stage 3: u = Zh[:512]^T @ v0 --------------------------------------
    {
        const int k = tid & 63, part = tid >> 6;   // 4 partial sums per column
        float s = 0.f;
        for (int m = part * 128; m < part * 128 + 128; ++m)
            s += Zb[(size_t)m * B_COLS + k] * v0[m];
        atomicAdd(&uvec[k], s);
    }
    __syncthreads();

    // ---- stage 4: 64-dim layer recursion -----------------------------------
    // t = gamma_i*t + (c_i/512)*u ; S += alpha_i*t ; u += alpha_i*(K t)
    for (int i = 0; i < NLAYER; ++i) {
        const float sc = cacc[i] * (1.0f / NSEQ);
        const float g  = gamma[i];
        const float al = alpha[i];
        if (tid < DH) {
            float tn = g * tvec[tid] + sc * uvec[tid];
            tvec[tid] = tn;
            Svec[tid] += al * tn;
        }
        __syncthreads();
        if (tid < DH) {
            float acc = 0.f;
            #pragma unroll 8
            for (int j = 0; j < DH; ++j) acc += Kmat[tid * DH + j] * tvec[j];
            ktv[tid] = acc;
        }
        __syncthreads();
        if (tid < DH) uvec[tid] += al * ktv[tid];
        __syncthreads();
    }

    // ---- stage 5: writeout (copy cols 0..63, col 64 = v0 + Zh@S) -----------
    for (int n = wave; n < B_ROWS; n += 8) {
        float z0 = Zb[(size_t)n * B_COLS + lane];        // cols 0..31
        float z1 = Zb[(size_t)n * B_COLS + 32 + lane];   // cols 32..63
        float d  = z0 * Svec[lane] + z1 * Svec[32 + lane];
        #pragma unroll
        for (int off = 16; off >= 1; off >>= 1) d += __shfl_xor(d, off, 32);
        Ob[(size_t)n * B_COLS + lane]      = z0;
        Ob[(size_t)n * B_COLS + 32 + lane] = z1;
        if (lane == 0) Ob[(size_t)n * B_COLS + DH] = v0[n] + d;
    }
}

extern "C" void kernel_launch(void* const* d_in, const int* in_sizes, int n_in,
                              void* d_out, int out_size, void* d_ws, size_t ws_size,
                              hipStream_t stream) {
    (void)n_in; (void)out_size; (void)d_ws; (void)ws_size;
    const float* Z        = (const float*)d_in[0];
    const float* allparam = (const float*)d_in[1];
    const float* gamma    = (const float*)d_in[2];
    const float* alpha    = (const float*)d_in[3];
    float* out = (float*)d_out;

    const int nb = in_sizes[0] / (B_ROWS * B_COLS);   // batch = 512
    tf_col64_kernel<<<nb, 256, 0, stream>>>(Z, allparam, gamma, alpha, out);
}